// LeakySNN_4870492914050
// MI455X (gfx1250) — compile-verified
//
#include <hip/hip_runtime.h>

// CDNA5 / gfx1250 fused Leaky-SNN forward:
//   x1 = x @ W1^T + b1  (f32 WMMA 16x16x4, K=256)
//   25-step leaky integrate-and-fire recurrence kept entirely in VGPRs
//   out = spk_sum @ W2^T + b2  (cross-lane + LDS reduction)

typedef __attribute__((ext_vector_type(2))) float v2f;
typedef __attribute__((ext_vector_type(8))) float v8f;

#define NI          256
#define NH          1024
#define TILE_M      16
#define NSTEPS      25
#define LDS_STRIDE  260   // 256 + 4 pad -> conflict-free ds_load_b64 across 16 lanes

__launch_bounds__(256)
__global__ void snn_fused_kernel(const float* __restrict__ x,
                                 const float* __restrict__ W1,
                                 const float* __restrict__ b1,
                                 const float* __restrict__ W2,
                                 const float* __restrict__ b2,
                                 const float* __restrict__ beta_p,
                                 const float* __restrict__ thr,
                                 float* __restrict__ out)
{
    __shared__ float xs[TILE_M * LDS_STRIDE];
    __shared__ float outs[TILE_M];

    const int tid  = threadIdx.x;
    const int lane = tid & 31;          // wave32
    const int wave = tid >> 5;          // 0..7
    const int l16  = lane & 15;         // column / row-in-tile index
    const int half = lane >> 4;         // 0: K+0..1 / rows 0-7 ; 1: K+2..3 / rows 8-15
    const int rowBase = blockIdx.x * TILE_M;

    if (tid < TILE_M) outs[tid] = 0.0f;

    // ---- cooperative stage of x tile [16 x 256] (rows are contiguous) into LDS
    {
        const float4* xg = reinterpret_cast<const float4*>(x + (size_t)rowBase * NI);
        #pragma unroll
        for (int j = 0; j < 4; ++j) {
            int i = tid + 256 * j;                  // float4 index 0..1023
            float4 v = xg[i];
            int m = i >> 6;                         // 64 float4 per row
            int c = (i & 63) << 2;
            *reinterpret_cast<float4*>(&xs[m * LDS_STRIDE + c]) = v;
        }
    }
    __syncthreads();

    const float beta_c = fminf(fmaxf(beta_p[0], 0.0f), 1.0f);

    float accOut[8];
    #pragma unroll
    for (int r = 0; r < 8; ++r) accOut[r] = 0.0f;

    // each wave owns 8 chunks of 16 N-columns: n in [wave*128, wave*128+128)
    for (int cidx = 0; cidx < 8; ++cidx) {
        const int nbase = wave * 128 + cidx * 16;
        const int n = nbase + l16;

        // ---- x1 tile = x @ W1^T + b1 via V_WMMA_F32_16X16X4_F32 over K=256
        const float b1n = b1[n];
        v8f acc;
        #pragma unroll
        for (int r = 0; r < 8; ++r) acc[r] = b1n;   // b1 broadcast per column n

        // A frag: lane half selects K pair; row = l16 (from LDS x tile)
        // B frag: mirrored layout; B[k][n] = W1[n][k] -> contiguous float2 per lane
        const float* xrow  = &xs[l16 * LDS_STRIDE + 2 * half];
        const float* w1row = W1 + (size_t)n * NI + 2 * half;

        #pragma unroll 8
        for (int k0 = 0; k0 < NI; k0 += 4) {
            v2f a = *reinterpret_cast<const v2f*>(xrow + k0);    // ds_load_b64
            v2f b = *reinterpret_cast<const v2f*>(w1row + k0);   // global b64, W1 L2-resident
            acc = __builtin_amdgcn_wmma_f32_16x16x4_f32(
                    false, a, false, b, (short)0, acc, false, false);
        }

        // ---- 25-step leaky integrate-and-fire, fully in registers.
        // acc element r on this lane = x1[row = r + 8*half][col = nbase + l16]
        float x1a[8], mem[8], ssum[8];
        #pragma unroll
        for (int r = 0; r < 8; ++r) { x1a[r] = acc[r]; mem[r] = 0.0f; ssum[r] = 0.0f; }

        const float thrv = thr[n];
        for (int s = 0; s < NSTEPS; ++s) {
            #pragma unroll
            for (int r = 0; r < 8; ++r) {
                float resetT = (mem[r] > thrv) ? thrv : 0.0f;      // reset*thr
                mem[r] = fmaf(beta_c, mem[r], x1a[r]) - resetT;
                ssum[r] += (mem[r] > thrv) ? 1.0f : 0.0f;          // spike accumulate
            }
        }

        // ---- fold spk_sum * W2[n] into per-row partial output
        const float w2n = W2[n];
        #pragma unroll
        for (int r = 0; r < 8; ++r) accOut[r] = fmaf(ssum[r], w2n, accOut[r]);
    }

    // ---- reduce over the 16 columns held by each 16-lane group
    #pragma unroll
    for (int r = 0; r < 8; ++r) {
        float v = accOut[r];
        v += __shfl_xor(v, 1, 32);
        v += __shfl_xor(v, 2, 32);
        v += __shfl_xor(v, 4, 32);
        v += __shfl_xor(v, 8, 32);
        accOut[r] = v;
    }
    if (l16 == 0) {                       // lane 0 -> rows 0..7, lane 16 -> rows 8..15
        #pragma unroll
        for (int r = 0; r < 8; ++r)
            atomicAdd(&outs[half * 8 + r], accOut[r]);   // ds_add_f32 across 8 waves
    }
    __syncthreads();

    if (tid < TILE_M) out[rowBase + tid] = outs[tid] + b2[0];
}

extern "C" void kernel_launch(void* const* d_in, const int* in_sizes, int n_in,
                              void* d_out, int out_size, void* d_ws, size_t ws_size,
                              hipStream_t stream) {
    (void)n_in; (void)out_size; (void)d_ws; (void)ws_size;
    const float* x    = (const float*)d_in[0];
    const float* W1   = (const float*)d_in[1];
    const float* b1   = (const float*)d_in[2];
    const float* W2   = (const float*)d_in[3];
    const float* b2   = (const float*)d_in[4];
    const float* beta = (const float*)d_in[5];
    const float* thr  = (const float*)d_in[6];
    float* out        = (float*)d_out;

    const int B = in_sizes[0] / NI;          // 32768
    dim3 grid(B / TILE_M), block(256);
    hipLaunchKernelGGL(snn_fused_kernel, grid, block, 0, stream,
                       x, W1, b1, W2, b2, beta, thr, out);
}